// Latent_40853728919873
// MI455X (gfx1250) — compile-verified
//
#include <hip/hip_runtime.h>
#include <stdint.h>

#define N_PTS 4096
#define M_PTS 1024
#define DIMS  64
#define TN    128               // z rows per LDS tile (32 KB)

typedef float v2f __attribute__((ext_vector_type(2)));
typedef float v8f __attribute__((ext_vector_type(8)));

// CDNA5 async global->LDS copy (ASYNCcnt-tracked). GV mode: 64-bit vaddr, saddr=off.
__device__ __forceinline__ void async_ld_b128(uint32_t lds_off, uint64_t gaddr) {
  asm volatile("global_load_async_to_lds_b128 %0, %1, off"
               :: "v"(lds_off), "v"(gaddr)
               : "memory");
}

__device__ __forceinline__ void wait_asynccnt0() {
  asm volatile("s_wait_asynccnt 0" ::: "memory");
}

// Kernel 1: per (m,d) min over n of (z[n,d]-e[m,d])^2 ; per-block partial sums.
__global__ __launch_bounds__(256)
void vq_min_kernel(const float* __restrict__ z, const float* __restrict__ e,
                   float* __restrict__ partials) {
  __shared__ float tile[TN * DIMS];   // 32 KB
  __shared__ float wsum[8];

  const int t      = threadIdx.x;
  const int d      = t & (DIMS - 1);
  const int mlocal = t >> 6;                       // 0..3
  const int m      = (blockIdx.x << 2) + mlocal;   // 4 m-rows per block

  const float ev = e[m * DIMS + d];
  float acc = __builtin_inff();

  const uint64_t zbase   = (uint64_t)(uintptr_t)z;
  const uint32_t ldsbase = (uint32_t)(uintptr_t)(void*)tile;

  for (int n0 = 0; n0 < N_PTS; n0 += TN) {
    // cooperative async copy of z[n0 .. n0+TN) x 64 f32 into LDS (2048 x 16B chunks)
    const uint64_t gtile = zbase + (uint64_t)(n0 * DIMS * 4);
    #pragma unroll
    for (int k = 0; k < (TN * DIMS * 4) / (16 * 256); ++k) {   // 8 chunks/thread
      const int chunk = t + (k << 8);
      async_ld_b128(ldsbase + chunk * 16, gtile + (uint64_t)chunk * 16);
    }
    wait_asynccnt0();
    __syncthreads();

    #pragma unroll 8
    for (int nl = 0; nl < TN; ++nl) {
      const float v    = tile[nl * DIMS + d];     // bank == d: conflict-free
      const float diff = v - ev;
      acc = fminf(acc, diff * diff);
    }
    __syncthreads();
  }

  // Wave-level exact sum of 32 'acc' values via WMMA against ones:
  //   A(16x4): lane i<16 -> A[i][0]=acc ; lane i+16 -> A[i][2]=acc ; rest 0
  //   D = A x Ones(4x16) + 0  =>  D[i][j] = acc_i + acc_{i+16}
  v2f a; a.x = acc;  a.y = 0.0f;
  v2f b; b.x = 1.0f; b.y = 1.0f;
  v8f c = {};
  v8f dm = __builtin_amdgcn_wmma_f32_16x16x4_f32(false, a, false, b,
                                                 (short)0, c, false, false);
  float part = dm[0] + dm[1] + dm[2] + dm[3] + dm[4] + dm[5] + dm[6] + dm[7];
  // Column N=0 of D: rows 0..7 live in lane 0, rows 8..15 in lane 16.
  float s0  = __builtin_bit_cast(float, __builtin_amdgcn_readlane(__builtin_bit_cast(int, part), 0));
  float s16 = __builtin_bit_cast(float, __builtin_amdgcn_readlane(__builtin_bit_cast(int, part), 16));
  const float wave_total = s0 + s16;

  const int wave = t >> 5;
  if ((t & 31) == 0) wsum[wave] = wave_total;
  __syncthreads();
  if (t == 0) {
    float s = 0.0f;
    #pragma unroll
    for (int w = 0; w < 8; ++w) s += wsum[w];
    partials[blockIdx.x] = s;
  }
}

// Kernel 2: deterministic fixed-order final reduction -> d_out[0]
__global__ void finalize_kernel(const float* __restrict__ partials,
                                float* __restrict__ out) {
  if (threadIdx.x == 0 && blockIdx.x == 0) {
    float s = 0.0f;
    for (int i = 0; i < M_PTS / 4; ++i) s += partials[i];
    out[0] = s * (1.0f / (float)(M_PTS * DIMS));
  }
}

// Kernel 3: z_masked and z_copy streams
__global__ __launch_bounds__(256)
void mask_copy_kernel(const float* __restrict__ z, const int* __restrict__ idx,
                      float* __restrict__ out_masked, float* __restrict__ out_copy) {
  const int i   = blockIdx.x * 256 + threadIdx.x;   // over N*D
  const int row = i >> 6;
  const int col = i & 63;
  const float v = z[i];
  out_masked[i] = (col < idx[row]) ? v : 0.0f;
  out_copy[i]   = v;
}

extern "C" void kernel_launch(void* const* d_in, const int* in_sizes, int n_in,
                              void* d_out, int out_size, void* d_ws, size_t ws_size,
                              hipStream_t stream) {
  const float* z    = (const float*)d_in[0];
  const float* e    = (const float*)d_in[1];
  const int*   ridx = (const int*)d_in[2];
  float*       out  = (float*)d_out;
  float*       part = (float*)d_ws;                 // 256 floats of scratch

  vq_min_kernel<<<M_PTS / 4, 256, 0, stream>>>(z, e, part);
  finalize_kernel<<<1, 32, 0, stream>>>(part, out);
  mask_copy_kernel<<<(N_PTS * DIMS) / 256, 256, 0, stream>>>(
      z, ridx, out + 1, out + 1 + N_PTS * DIMS);
}